// BatchedExternalMemory_53523882443200
// MI455X (gfx1250) — compile-verified
//
#include <hip/hip_runtime.h>
#include <math.h>

#define NB      32768
#define DIM     128
#define NSLOT   16
#define NAGENTS 8192
#define BT      16     // batch rows per block
#define AS      132    // LDS row stride (floats) to dodge bank conflicts

typedef __attribute__((ext_vector_type(2))) float v2f;
typedef __attribute__((ext_vector_type(8))) float v8f;

__device__ __forceinline__ float red8(float v) {
    v += __shfl_xor(v, 1);
    v += __shfl_xor(v, 2);
    v += __shfl_xor(v, 4);
    return v;
}
__device__ __forceinline__ float red32(float v) {
    v += __shfl_xor(v, 1);
    v += __shfl_xor(v, 2);
    v += __shfl_xor(v, 4);
    v += __shfl_xor(v, 8);
    v += __shfl_xor(v, 16);
    return v;
}

enum { ACT_NONE = 0, ACT_GELU = 1, ACT_SIG = 2, ACT_TANH = 3 };

// Out[16 x 128] = A[16 x 128] @ W^T + bias, optional activation.
// A, Out are LDS tiles with row stride AS. W is [128 out][128 in] row-major in global.
// 128 threads = 4 waves; wave w computes output column tiles [32w, 32w+32).
// f32 WMMA 16x16x4: A frag lane l -> M = l%16, VGPR v -> K = (l/16)*2 + v.
// B frag (K x N, B[k][n] = W[n][k]) mirrored: lane l -> N = l%16, VGPR v -> K = (l/16)*2 + v.
// C/D frag: lane l, VGPR r -> M = r + (l/16)*8, N = l%16.
template<int ACT>
__device__ __forceinline__ void gemm16(const float* __restrict__ W,
                                       const float* __restrict__ bias,
                                       const float* __restrict__ A,
                                       float* __restrict__ Out,
                                       int tid)
{
    const int lane = tid & 31;
    const int wv   = tid >> 5;
    const int m    = lane & 15;
    const int kh   = (lane >> 4) << 1;   // 0 or 2
    const int col  = lane & 15;
    const int n0   = wv * 32;
    const int n1   = n0 + 16;
    const float* __restrict__ w0 = W + (size_t)(n0 + col) * DIM;
    const float* __restrict__ w1 = W + (size_t)(n1 + col) * DIM;
    const float* __restrict__ ar = A + m * AS;

    v8f c0 = {0.f, 0.f, 0.f, 0.f, 0.f, 0.f, 0.f, 0.f};
    v8f c1 = {0.f, 0.f, 0.f, 0.f, 0.f, 0.f, 0.f, 0.f};

    #pragma unroll 8
    for (int kk = 0; kk < DIM; kk += 4) {
        const int k = kk + kh;
        v2f av = *(const v2f*)(ar + k);          // 8B aligned: AS even, k even
        v2f b0; b0.x = w0[k]; b0.y = w0[k + 1];
        v2f b1; b1.x = w1[k]; b1.y = w1[k + 1];
        c0 = __builtin_amdgcn_wmma_f32_16x16x4_f32(false, av, false, b0, (short)0, c0, false, false);
        c1 = __builtin_amdgcn_wmma_f32_16x16x4_f32(false, av, false, b1, (short)0, c1, false, false);
    }

    const int rh  = (lane >> 4) << 3;            // 0 or 8
    const float bb0 = bias[n0 + col];
    const float bb1 = bias[n1 + col];
    #pragma unroll
    for (int r = 0; r < 8; ++r) {
        float v0 = c0[r] + bb0;
        float v1 = c1[r] + bb1;
        if (ACT == ACT_GELU) {
            v0 = 0.5f * v0 * (1.f + erff(v0 * 0.70710678118654752f));
            v1 = 0.5f * v1 * (1.f + erff(v1 * 0.70710678118654752f));
        } else if (ACT == ACT_SIG) {
            v0 = 1.f / (1.f + expf(-v0));
            v1 = 1.f / (1.f + expf(-v1));
        } else if (ACT == ACT_TANH) {
            v0 = tanhf(v0);
            v1 = tanhf(v1);
        }
        Out[(r + rh) * AS + n0 + col] = v0;
        Out[(r + rh) * AS + n1 + col] = v1;
    }
}

__device__ __forceinline__ void load_tile(const float* __restrict__ src,
                                          float* __restrict__ dst, int tid)
{
    const float4* s4 = (const float4*)src;
    for (int i = tid; i < BT * DIM / 4; i += 128) {
        float4 v = s4[i];
        const int row = i >> 5;            // 32 float4 per row
        const int c   = (i & 31) << 2;
        float* d = dst + row * AS + c;
        d[0] = v.x; d[1] = v.y; d[2] = v.z; d[3] = v.w;
    }
}

// ---------------------------------------------------------------------------
// Read path: pq = Q@Wq^T+bq ; cosine attention over slots ; h = gelu(read@Wf^T+bf);
// LayerNorm(h)*gamma+beta  -> out_read[B][128]
// ---------------------------------------------------------------------------
__global__ __launch_bounds__(128) void read_kernel(
    const float* __restrict__ queries, const int* __restrict__ agent_idx,
    const float* __restrict__ memory,
    const float* __restrict__ Wq, const float* __restrict__ bq,
    const float* __restrict__ Wf, const float* __restrict__ bf,
    const float* __restrict__ gamma, const float* __restrict__ beta,
    float* __restrict__ out)
{
    __shared__ float Qs[BT * AS];
    __shared__ float Ps[BT * AS];
    __shared__ float Rs[BT * AS];
    const int tid = threadIdx.x;
    const int rowbase = blockIdx.x * BT;

    load_tile(queries + (size_t)rowbase * DIM, Qs, tid);
    __syncthreads();

    gemm16<ACT_NONE>(Wq, bq, Qs, Ps, tid);      // Ps = projected queries
    __syncthreads();

    // 8 threads per batch row; each handles 16 dims.
    const int r = tid >> 3;
    const int g = tid & 7;
    const int a = agent_idx[rowbase + r];
    const float* __restrict__ mrow = memory + (size_t)a * (NSLOT * DIM);

    float p[16];
    float ssq = 0.f;
    #pragma unroll
    for (int j = 0; j < 16; ++j) { p[j] = Ps[r * AS + g * 16 + j]; ssq += p[j] * p[j]; }
    ssq = red8(ssq);
    const float pn = fmaxf(sqrtf(ssq), 1e-12f);

    float sc[NSLOT];
    #pragma unroll
    for (int s = 0; s < NSLOT; ++s) {
        const float* m = mrow + s * DIM + g * 16;
        float dot = 0.f, msq = 0.f;
        #pragma unroll
        for (int j = 0; j < 16; ++j) { float mv = m[j]; dot += p[j] * mv; msq += mv * mv; }
        dot = red8(dot);
        msq = red8(msq);
        const float mn = fmaxf(sqrtf(msq), 1e-12f);
        sc[s] = dot / (pn * mn) * 0.08838834764831845f;   // 1/sqrt(128)
    }

    float mx = sc[0];
    #pragma unroll
    for (int s = 1; s < NSLOT; ++s) mx = fmaxf(mx, sc[s]);
    float sum = 0.f;
    #pragma unroll
    for (int s = 0; s < NSLOT; ++s) { sc[s] = expf(sc[s] - mx); sum += sc[s]; }
    const float inv = 1.f / sum;

    float acc[16];
    #pragma unroll
    for (int j = 0; j < 16; ++j) acc[j] = 0.f;
    #pragma unroll
    for (int s = 0; s < NSLOT; ++s) {
        const float w = sc[s] * inv;
        const float* m = mrow + s * DIM + g * 16;
        #pragma unroll
        for (int j = 0; j < 16; ++j) acc[j] += w * m[j];
    }
    #pragma unroll
    for (int j = 0; j < 16; ++j) Rs[r * AS + g * 16 + j] = acc[j];
    __syncthreads();

    gemm16<ACT_GELU>(Wf, bf, Rs, Qs, tid);      // Qs reused as h
    __syncthreads();

    float h[16];
    float sm = 0.f, s2 = 0.f;
    #pragma unroll
    for (int j = 0; j < 16; ++j) {
        h[j] = Qs[r * AS + g * 16 + j];
        sm += h[j]; s2 += h[j] * h[j];
    }
    sm = red8(sm);
    s2 = red8(s2);
    const float mu   = sm * (1.f / 128.f);
    const float var  = s2 * (1.f / 128.f) - mu * mu;
    const float rstd = rsqrtf(var + 1e-5f);
    float* orow = out + (size_t)(rowbase + r) * DIM + g * 16;
    #pragma unroll
    for (int j = 0; j < 16; ++j) {
        const int d = g * 16 + j;
        orow[j] = (h[j] - mu) * rstd * gamma[d] + beta[d];
    }
}

// ---------------------------------------------------------------------------
// Write path pass 1: projections + softmax weights + atomic scatter of
// w[s]*erase*intensity and w[s]*add*intensity into per-agent accumulators.
// ---------------------------------------------------------------------------
__global__ __launch_bounds__(128) void write_kernel(
    const float* __restrict__ wqueries, const int* __restrict__ widx,
    const float* __restrict__ memory,
    const float* __restrict__ Wq, const float* __restrict__ bq,
    const float* __restrict__ We, const float* __restrict__ be,
    const float* __restrict__ Wa, const float* __restrict__ ba,
    const float* __restrict__ Wg, const float* __restrict__ bg,
    float* __restrict__ Ebuf, float* __restrict__ Abuf, int* __restrict__ flags)
{
    __shared__ float Qs[BT * AS];
    __shared__ float Ps[BT * AS];
    __shared__ float Es[BT * AS];
    __shared__ float Aw[BT * AS];
    const int tid = threadIdx.x;
    const int rowbase = blockIdx.x * BT;

    load_tile(wqueries + (size_t)rowbase * DIM, Qs, tid);
    __syncthreads();

    gemm16<ACT_NONE>(Wq, bq, Qs, Ps, tid);      // projected write queries
    gemm16<ACT_SIG >(We, be, Qs, Es, tid);      // erase gate
    gemm16<ACT_TANH>(Wa, ba, Qs, Aw, tid);      // add vector
    __syncthreads();

    const int r = tid >> 3;
    const int g = tid & 7;
    const int a = widx[rowbase + r];
    const float* __restrict__ mrow = memory + (size_t)a * (NSLOT * DIM);

    // intensity = sigmoid(q . Wg + bg)
    float gs = 0.f;
    #pragma unroll
    for (int j = 0; j < 16; ++j) gs += Qs[r * AS + g * 16 + j] * Wg[g * 16 + j];
    gs = red8(gs);
    const float inten = 1.f / (1.f + expf(-(gs + bg[0])));

    float p[16];
    float ssq = 0.f;
    #pragma unroll
    for (int j = 0; j < 16; ++j) { p[j] = Ps[r * AS + g * 16 + j]; ssq += p[j] * p[j]; }
    ssq = red8(ssq);
    const float pn = fmaxf(sqrtf(ssq), 1e-12f);

    float sc[NSLOT];
    #pragma unroll
    for (int s = 0; s < NSLOT; ++s) {
        const float* m = mrow + s * DIM + g * 16;
        float dot = 0.f, msq = 0.f;
        #pragma unroll
        for (int j = 0; j < 16; ++j) { float mv = m[j]; dot += p[j] * mv; msq += mv * mv; }
        dot = red8(dot);
        msq = red8(msq);
        const float mn = fmaxf(sqrtf(msq), 1e-12f);
        sc[s] = dot / (pn * mn) * 0.08838834764831845f;
    }
    float mx = sc[0];
    #pragma unroll
    for (int s = 1; s < NSLOT; ++s) mx = fmaxf(mx, sc[s]);
    float sum = 0.f;
    #pragma unroll
    for (int s = 0; s < NSLOT; ++s) { sc[s] = expf(sc[s] - mx); sum += sc[s]; }
    const float inv = 1.f / sum;

    float ev[16], av[16];
    #pragma unroll
    for (int j = 0; j < 16; ++j) {
        ev[j] = Es[r * AS + g * 16 + j] * inten;
        av[j] = Aw[r * AS + g * 16 + j] * inten;
    }

    const size_t abase = (size_t)a * (NSLOT * DIM) + g * 16;
    #pragma unroll
    for (int s = 0; s < NSLOT; ++s) {
        const float w = sc[s] * inv;
        float* ep = Ebuf + abase + s * DIM;
        float* ap = Abuf + abase + s * DIM;
        #pragma unroll
        for (int j = 0; j < 16; ++j) {
            atomicAdd(&ep[j], w * ev[j]);
            atomicAdd(&ap[j], w * av[j]);
        }
    }
    if (g == 0) flags[a] = 1;                   // idempotent presence mark
}

// ---------------------------------------------------------------------------
// Write path pass 2: new_mem = has_writer ? l2norm(mem*(1-E)+A) : mem
// One wave per (agent, slot) row of 128 dims.
// ---------------------------------------------------------------------------
__global__ __launch_bounds__(256) void finalize_kernel(
    const float* __restrict__ memory, const float* __restrict__ Ebuf,
    const float* __restrict__ Abuf, const int* __restrict__ flags,
    float* __restrict__ out_mem)
{
    const int wid  = (int)((blockIdx.x * 256 + threadIdx.x) >> 5);  // (a*16 + s)
    const int lane = threadIdx.x & 31;
    const int a    = wid >> 4;
    const size_t base = (size_t)wid * DIM;

    float4 mv = ((const float4*)(memory + base))[lane];
    float4 res;
    if (flags[a]) {
        float4 e = ((const float4*)(Ebuf + base))[lane];
        float4 ad = ((const float4*)(Abuf + base))[lane];
        float4 u;
        u.x = mv.x * (1.f - e.x) + ad.x;
        u.y = mv.y * (1.f - e.y) + ad.y;
        u.z = mv.z * (1.f - e.z) + ad.z;
        u.w = mv.w * (1.f - e.w) + ad.w;
        float ss = u.x * u.x + u.y * u.y + u.z * u.z + u.w * u.w;
        ss = red32(ss);
        const float invn = 1.f / fmaxf(sqrtf(ss), 1e-12f);
        res.x = u.x * invn; res.y = u.y * invn; res.z = u.z * invn; res.w = u.w * invn;
    } else {
        res = mv;
    }
    ((float4*)(out_mem + base))[lane] = res;
}

extern "C" void kernel_launch(void* const* d_in, const int* in_sizes, int n_in,
                              void* d_out, int out_size, void* d_ws, size_t ws_size,
                              hipStream_t stream) {
    const float* queries  = (const float*)d_in[0];
    const float* wqueries = (const float*)d_in[1];
    const int*   aidx     = (const int*)d_in[2];
    const int*   widx     = (const int*)d_in[3];
    const float* memory   = (const float*)d_in[4];
    const float* Wq       = (const float*)d_in[5];
    const float* bq       = (const float*)d_in[6];
    const float* We       = (const float*)d_in[7];
    const float* be       = (const float*)d_in[8];
    const float* Wa       = (const float*)d_in[9];
    const float* ba       = (const float*)d_in[10];
    const float* Wg       = (const float*)d_in[11];
    const float* bg       = (const float*)d_in[12];
    const float* Wf       = (const float*)d_in[13];
    const float* bf       = (const float*)d_in[14];
    const float* gamma    = (const float*)d_in[15];
    const float* beta     = (const float*)d_in[16];

    float* out_read = (float*)d_out;
    float* out_mem  = out_read + (size_t)NB * DIM;

    const size_t memElems = (size_t)NAGENTS * NSLOT * DIM;
    float* Ebuf = (float*)d_ws;
    float* Abuf = Ebuf + memElems;
    int*   flags = (int*)(Abuf + memElems);

    const size_t clearBytes = memElems * sizeof(float) * 2 + (size_t)NAGENTS * sizeof(int);
    hipMemsetAsync(d_ws, 0, clearBytes, stream);

    read_kernel<<<NB / BT, 128, 0, stream>>>(queries, aidx, memory, Wq, bq, Wf, bf,
                                             gamma, beta, out_read);
    write_kernel<<<NB / BT, 128, 0, stream>>>(wqueries, widx, memory, Wq, bq, We, be,
                                              Wa, ba, Wg, bg, Ebuf, Abuf, flags);
    finalize_kernel<<<(NAGENTS * NSLOT) / 8, 256, 0, stream>>>(memory, Ebuf, Abuf,
                                                               flags, out_mem);
}